// SAGE_1168231105073
// MI455X (gfx1250) — compile-verified
//
#include <hip/hip_runtime.h>
#include <hip/hip_bf16.h>

// ---------------------------------------------------------------------------
// GraphSAGE 3-layer forward for MI455X (gfx1250, wave32).
//  - Edge aggregation: per-edge wave32 scatter with global_atomic_add_f32
//    (features are L2-resident: 25.6MB << 192MB L2).
//  - Dense part: fused  H = relu( X*Wself + (NS*inv_deg)*Wneigh + b )
//    with V_WMMA_F32_16X16X4_F32 (exact fp32 matrix path on CDNA5).
//    Weights are pre-transposed so every B fragment is one b64 load with an
//    immediate offset, and B fragments are register-resident across a
//    grid-stride row-tile loop (weight reads amortized over ~6 tiles/wave).
// ---------------------------------------------------------------------------

typedef float v2f __attribute__((ext_vector_type(2)));
typedef float v8f __attribute__((ext_vector_type(8)));

#define K_DIM 128   // feature depth of every layer input (IN == HID == 128)

// ---------------- degree accumulation -------------------------------------
__global__ void sage_deg_kernel(const int* __restrict__ dst, float* __restrict__ deg, int nE) {
    int i = blockIdx.x * blockDim.x + threadIdx.x;
    if (i < nE) atomicAdd(&deg[dst[i]], 1.0f);
}

// deg -> 1/max(deg,1)  (in place)
__global__ void sage_invdeg_kernel(float* __restrict__ deg, int nN) {
    int i = blockIdx.x * blockDim.x + threadIdx.x;
    if (i < nN) deg[i] = 1.0f / fmaxf(deg[i], 1.0f);
}

// ---------------- weight transpose: W[K x N] -> Wt[N x K] ------------------
__global__ void sage_transpose_kernel(const float* __restrict__ W, float* __restrict__ Wt, int N) {
    int i = blockIdx.x * blockDim.x + threadIdx.x;
    if (i < K_DIM * N) {
        int k = i / N;
        int n = i - k * N;
        Wt[(size_t)n * K_DIM + k] = W[(size_t)k * N + n];
    }
}

// ---------------- edge scatter: NS[dst] += X[src] --------------------------
// one wave32 per edge; each lane loads a float4 (16B) -> full 512B row per wave
__global__ __launch_bounds__(256)
void sage_scatter_kernel(const float* __restrict__ X, const int* __restrict__ src,
                         const int* __restrict__ dst, float* __restrict__ NS, int nE) {
    const int wave = threadIdx.x >> 5;
    const int lane = threadIdx.x & 31;
    const int e = blockIdx.x * 8 + wave;
    if (e >= nE) return;
    const int s = src[e];
    const int d = dst[e];
    const float4* xr = (const float4*)(X + (size_t)s * K_DIM);
    float4 v = xr[lane];
    float* o = NS + (size_t)d * K_DIM + lane * 4;
    atomicAdd(o + 0, v.x);
    atomicAdd(o + 1, v.y);
    atomicAdd(o + 2, v.z);
    atomicAdd(o + 3, v.w);
}

// ---------------- fused WMMA GEMM layer ------------------------------------
//   H[r,c] = relu?( sum_k X[r,k]*Wself[k,c] + inv[r]*NS[r,k]*Wneigh[k,c] + b[c] )
// Block = (N/16) waves; wave w owns column tile w, B fragments live in VGPRs,
// grid-stride loop over 16-row tiles.
template <int N, bool RELU>
__global__ __launch_bounds__(N * 2)
void sage_wmma_gemm(const float* __restrict__ X, const float* __restrict__ NS,
                    const float* __restrict__ invdeg,
                    const float* __restrict__ WtS, const float* __restrict__ WtN,
                    const float* __restrict__ bias, float* __restrict__ H, int nRows) {
    const int lane = threadIdx.x & 31;
    const int wave = threadIdx.x >> 5;          // == column tile index
    const int c0 = wave << 4;
    const int m  = lane & 15;                   // A-row / B-col / D-col for this lane
    const int kh = lane >> 4;                   // K-half select within a K=4 step
    const int col = c0 + m;

    // Preload all B fragments (transposed weights: Wt[col][k], contiguous in k).
    // lane holds {K=4*ks+2*kh, K=4*ks+2*kh+1} of column `col`.
    v2f bs[32], bn[32];
    const float* wsp = WtS + (size_t)col * K_DIM + 2 * kh;
    const float* wnp = WtN + (size_t)col * K_DIM + 2 * kh;
#pragma unroll
    for (int ks = 0; ks < 32; ++ks) {
        bs[ks] = *(const v2f*)(wsp + 4 * ks);
        bn[ks] = *(const v2f*)(wnp + 4 * ks);
    }
    const float bv = bias[col];

    const int rowTiles = nRows >> 4;            // 3125 (nRows % 16 == 0)
    for (int rt = blockIdx.x; rt < rowTiles; rt += gridDim.x) {
        const int r0 = rt << 4;
        const float s = invdeg[r0 + m];         // per-row mean scale, lane-constant
        const float* xrow = X  + (size_t)(r0 + m) * K_DIM + 2 * kh;
        const float* nrow = NS + (size_t)(r0 + m) * K_DIM + 2 * kh;

        v8f acc = {0.f, 0.f, 0.f, 0.f, 0.f, 0.f, 0.f, 0.f};
#pragma unroll
        for (int ks = 0; ks < 32; ++ks) {
            v2f a_s = *(const v2f*)(xrow + 4 * ks);
            v2f a_n = *(const v2f*)(nrow + 4 * ks);
            a_n = a_n * s;                       // fold mean-divide into A operand
            acc = __builtin_amdgcn_wmma_f32_16x16x4_f32(false, a_s, false, bs[ks],
                                                        (short)0, acc, false, false);
            acc = __builtin_amdgcn_wmma_f32_16x16x4_f32(false, a_n, false, bn[ks],
                                                        (short)0, acc, false, false);
        }

        // Epilogue: C/D layout -> VGPR r holds M = r + 8*kh, col = c0 + m
        float* orow = H + (size_t)r0 * N + col;
#pragma unroll
        for (int r = 0; r < 8; ++r) {
            float v = acc[r] + bv;
            if (RELU) v = fmaxf(v, 0.0f);
            orow[(size_t)(r + 8 * kh) * N] = v;
        }
    }
}

// ---------------------------------------------------------------------------
extern "C" void kernel_launch(void* const* d_in, const int* in_sizes, int n_in,
                              void* d_out, int out_size, void* d_ws, size_t ws_size,
                              hipStream_t stream) {
    const float* X   = (const float*)d_in[0];
    const int*   src = (const int*)d_in[1];
    const int*   dst = (const int*)d_in[2];
    const float* Ws1 = (const float*)d_in[3];
    const float* Wn1 = (const float*)d_in[4];
    const float* b1  = (const float*)d_in[5];
    const float* Ws2 = (const float*)d_in[6];
    const float* Wn2 = (const float*)d_in[7];
    const float* b2  = (const float*)d_in[8];
    const float* Ws3 = (const float*)d_in[9];
    const float* Wn3 = (const float*)d_in[10];
    const float* b3  = (const float*)d_in[11];
    float* out = (float*)d_out;

    const int nN = in_sizes[0] / K_DIM;   // 50000
    const int nE = in_sizes[1];           // 640000

    char* ws = (char*)d_ws;
    const size_t featBytes = (size_t)nN * K_DIM * sizeof(float);   // 25.6 MB
    float* NS  = (float*)(ws);                                     // neighbor-sum accumulator
    float* H1  = (float*)(ws + featBytes);
    float* H2  = (float*)(ws + 2 * featBytes);
    float* deg = (float*)(ws + 3 * featBytes);                     // becomes inv-deg
    float* Wt  = (float*)(ws + 3 * featBytes + ((size_t)nN * sizeof(float) + 255 & ~(size_t)255));
    float* Wt1s = Wt;                    // 128x128 each
    float* Wt1n = Wt1s + 128 * 128;
    float* Wt2s = Wt1n + 128 * 128;
    float* Wt2n = Wt2s + 128 * 128;
    float* Wt3s = Wt2n + 128 * 128;      // 64x128 each
    float* Wt3n = Wt3s + 64 * 128;

    const int scatterBlocks = (nE + 7) / 8;

    // one-time prep: degree -> inverse degree; weight transposes
    hipMemsetAsync(deg, 0, (size_t)nN * sizeof(float), stream);
    sage_deg_kernel<<<(nE + 255) / 256, 256, 0, stream>>>(dst, deg, nE);
    sage_invdeg_kernel<<<(nN + 255) / 256, 256, 0, stream>>>(deg, nN);
    sage_transpose_kernel<<<(128 * 128 + 255) / 256, 256, 0, stream>>>(Ws1, Wt1s, 128);
    sage_transpose_kernel<<<(128 * 128 + 255) / 256, 256, 0, stream>>>(Wn1, Wt1n, 128);
    sage_transpose_kernel<<<(128 * 128 + 255) / 256, 256, 0, stream>>>(Ws2, Wt2s, 128);
    sage_transpose_kernel<<<(128 * 128 + 255) / 256, 256, 0, stream>>>(Wn2, Wt2n, 128);
    sage_transpose_kernel<<<(128 * 64 + 255) / 256, 256, 0, stream>>>(Ws3, Wt3s, 64);
    sage_transpose_kernel<<<(128 * 64 + 255) / 256, 256, 0, stream>>>(Wn3, Wt3n, 64);

    // ---- layer 1 ----
    hipMemsetAsync(NS, 0, featBytes, stream);
    sage_scatter_kernel<<<scatterBlocks, 256, 0, stream>>>(X, src, dst, NS, nE);
    sage_wmma_gemm<128, true><<<512, 256, 0, stream>>>(X, NS, deg, Wt1s, Wt1n, b1, H1, nN);

    // ---- layer 2 ----
    hipMemsetAsync(NS, 0, featBytes, stream);
    sage_scatter_kernel<<<scatterBlocks, 256, 0, stream>>>(H1, src, dst, NS, nE);
    sage_wmma_gemm<128, true><<<512, 256, 0, stream>>>(H1, NS, deg, Wt2s, Wt2n, b2, H2, nN);

    // ---- layer 3 (N = 64, no relu) ----
    hipMemsetAsync(NS, 0, featBytes, stream);
    sage_scatter_kernel<<<scatterBlocks, 256, 0, stream>>>(H2, src, dst, NS, nE);
    sage_wmma_gemm<64, false><<<768, 128, 0, stream>>>(H2, NS, deg, Wt3s, Wt3n, b3, out, nN);
}